// MyGaussianConv_15762529976746
// MI455X (gfx1250) — compile-verified
//
#include <hip/hip_runtime.h>
#include <math.h>

// CDNA5 / gfx1250: fused separable 15x15 Gaussian blur, reflect padding.
// fp32 matrix path: V_WMMA_F32_16X16X4_F32 (wave32).
//
// One wave = one 16-wide x 64-tall output strip:
//   stage  : 80x32 input region -> LDS (coalesced b128 loads; border strips
//            take a uniform-branch reflect slow path)
//   pass 1 : T(80x16) = In(80x32) x Wh(32x16): 5 row-blocks x 8 K-chunks
//            = 40 WMMAs; T stored transposed via ds_store_b128
//   pass 2 : 4x Out(16x16) = Wv(16x32) x T-slice(32x16): 32 WMMAs
// Banded weight fragments (Wh/Wv pre-swizzled into the WMMA A/B register
// layout) are built once in d_ws by a prep kernel; each wave fetches them
// with 8 b128 loads from the L0-resident 4KB table.

typedef __attribute__((ext_vector_type(2))) float v2f;
typedef __attribute__((ext_vector_type(4))) float v4f;
typedef __attribute__((ext_vector_type(8))) float v8f;

#define KS     15
#define SROWS  80    // staged input rows per strip (64 + 14 halo, rounded to 80)
#define SSTR   36    // stage row stride (floats): 36m mod 64 = perm of 4Z -> conflict-free
#define TSTR   84    // T row stride (floats): 84n mod 64 = perm of 4Z -> conflict-free

__device__ __forceinline__ int reflect_idx(int i, int n) {
  i = (i < 0) ? -i : i;
  return (i >= n) ? (2 * n - 2 - i) : i;
}

// ---------------- prep: build banded weight fragments (lane-major) ---------
// ws[lane*16 + kk*2 + v]        : Wh fragment  Wh[c][n] = wx[c-n-1]
// ws[512 + lane*16 + kk*2 + v]  : Wv fragment  Wv[m][k] = wy[k-m]
__global__ __launch_bounds__(256)
void gauss15_prep_kernel(const float* __restrict__ k2d, float* __restrict__ wsf)
{
  __shared__ float w[32];   // [0..14]=wx (col sums), [16..30]=wy (row sums)
  const int t = threadIdx.x;
  if (t < KS) {
    float s = 0.0f;
    #pragma unroll
    for (int i = 0; i < KS; ++i) s += k2d[i * KS + t];
    w[t] = s;
  } else if (t >= 16 && t < 16 + KS) {
    const int r = t - 16;
    float s = 0.0f;
    #pragma unroll
    for (int j = 0; j < KS; ++j) s += k2d[r * KS + j];
    w[t] = s;
  }
  __syncthreads();

  const int kk   = t >> 5;      // 0..7
  const int lane = t & 31;
  const int n    = lane & 15;
  const int h    = lane >> 4;
  #pragma unroll
  for (int v = 0; v < 2; ++v) {
    const int c  = 4 * kk + 2 * h + v;   // K index in A/B fragment layout
    const int wi = c - n - 1;            // horizontal band (shifted origin)
    const int vi = c - n;                // vertical band
    wsf[lane * 16 + kk * 2 + v]       = (wi >= 0 && wi < KS) ? w[wi] : 0.0f;
    wsf[512 + lane * 16 + kk * 2 + v] = (vi >= 0 && vi < KS) ? w[16 + vi] : 0.0f;
  }
}

// ---------------- main: one wave per 16x64 output strip ---------------------
__global__ __launch_bounds__(32)
void gauss15_wmma_f32_kernel(const float* __restrict__ img,
                             const float* __restrict__ wsf,
                             float* __restrict__ out,
                             int H, int W)
{
  __shared__ float S[SROWS * SSTR];   // staged input region [80][32] (stride 36)
  __shared__ float T[16 * TSTR];      // transposed intermediate [n=16][r=80] (stride 84)

  const int lane = threadIdx.x;       // 0..31 (wave32)
  const int n    = lane & 15;
  const int half = lane >> 4;

  const int tx  = blockIdx.x;
  const int ty  = blockIdx.y;
  const int gx0 = tx * 16 - 8;        // col origin (4-float aligned); col c <-> gx0+c
  const int gy0 = ty * 64 - 7;        // row origin; row r <-> gy0+r

  // Weight fragments: 8x b128 from the 4KB lane-major table (L0-resident).
  v2f whf[8], wvf[8];
  {
    const v4f* wp = (const v4f*)(wsf + lane * 16);
    const v4f* vp = (const v4f*)(wsf + 512 + lane * 16);
    #pragma unroll
    for (int p = 0; p < 4; ++p) {
      const v4f qh = wp[p];
      const v4f qv = vp[p];
      whf[2 * p]     = (v2f){ qh[0], qh[1] };
      whf[2 * p + 1] = (v2f){ qh[2], qh[3] };
      wvf[2 * p]     = (v2f){ qv[0], qv[1] };
      wvf[2 * p + 1] = (v2f){ qv[2], qv[3] };
    }
  }

  // ---- stage input region into LDS ----
  const int lr = lane >> 3;           // row offset within 4-row group
  const int c4 = (lane & 7) * 4;      // fixed column (float4)
  const bool interior = (gx0 >= 0) && (gx0 + 32 <= W) &&
                        (gy0 >= 0) && (gy0 + SROWS <= H);
  if (interior) {
    const float* p = img + (size_t)(gy0 + lr) * W + (gx0 + c4);
    const int sbase = lr * SSTR + c4;
    #pragma unroll
    for (int j = 0; j < 20; ++j) {    // rows advance 4 per step: 20*4 = 80 rows
      *(v4f*)&S[sbase + j * 4 * SSTR] = *(const v4f*)p;
      p += (size_t)4 * W;
    }
  } else {
    #pragma unroll 1
    for (int j = 0; j < 20; ++j) {
      const int row = 4 * j + lr;
      const int gy  = reflect_idx(gy0 + row, H);
      const float* rp = img + (size_t)gy * W;
      v4f val;
      #pragma unroll
      for (int u = 0; u < 4; ++u) val[u] = rp[reflect_idx(gx0 + c4 + u, W)];
      *(v4f*)&S[row * SSTR + c4] = val;
    }
  }
  __syncthreads();

  // ---- pass 1: horizontal, T = In x Wh (5 row-blocks of 16) ----
  #pragma unroll
  for (int blk = 0; blk < 5; ++blk) {
    v8f acc = {};
    #pragma unroll
    for (int kk = 0; kk < 8; ++kk) {
      const int c = 4 * kk + 2 * half;                      // even -> b64 aligned
      const v2f a = *(const v2f*)&S[(blk * 16 + n) * SSTR + c];
      acc = __builtin_amdgcn_wmma_f32_16x16x4_f32(false, a, false, whf[kk],
                                                  (short)0, acc, false, false);
    }
    // D layout row = v + 8*half  ->  T[n][blk*16 + v + 8*half], two b128 stores
    const v4f lo = { acc[0], acc[1], acc[2], acc[3] };
    const v4f hi = { acc[4], acc[5], acc[6], acc[7] };
    *(v4f*)&T[n * TSTR + blk * 16 + 8 * half]     = lo;
    *(v4f*)&T[n * TSTR + blk * 16 + 8 * half + 4] = hi;
  }
  __syncthreads();

  // ---- pass 2: vertical, Out = Wv x T-slice (4 output tiles) ----
  const int  ox   = tx * 16 + n;
  const bool full = (tx * 16 + 16 <= W) && (ty * 64 + 64 <= H);
  #pragma unroll
  for (int t = 0; t < 4; ++t) {
    v8f acc = {};
    #pragma unroll
    for (int kk = 0; kk < 8; ++kk) {
      const int k = t * 16 + 4 * kk + 2 * half;             // even -> b64 aligned
      const v2f b = *(const v2f*)&T[n * TSTR + k];
      acc = __builtin_amdgcn_wmma_f32_16x16x4_f32(false, wvf[kk], false, b,
                                                  (short)0, acc, false, false);
    }
    const int oy0 = ty * 64 + t * 16 + 8 * half;            // D row = v + 8*half
    if (full) {
      float* op = out + (size_t)oy0 * W + ox;
      #pragma unroll
      for (int v = 0; v < 8; ++v) {
        *op = acc[v];
        op += W;
      }
    } else {
      #pragma unroll
      for (int v = 0; v < 8; ++v) {
        const int oy = oy0 + v;
        if (oy < H && ox < W) out[(size_t)oy * W + ox] = acc[v];
      }
    }
  }
}

extern "C" void kernel_launch(void* const* d_in, const int* in_sizes, int n_in,
                              void* d_out, int out_size, void* d_ws, size_t ws_size,
                              hipStream_t stream) {
  (void)n_in; (void)ws_size; (void)out_size;
  const float* img = (const float*)d_in[0];   // [H, W] fp32
  const float* k2d = (const float*)d_in[1];   // [15, 15] fp32
  float* out = (float*)d_out;                 // [H, W] fp32 (stride == 1)
  float* wsf = (float*)d_ws;                  // 1024 floats of weight fragments

  const int side = (int)llround(sqrt((double)in_sizes[0]));  // 8192
  const int H = side, W = side;

  gauss15_prep_kernel<<<1, 256, 0, stream>>>(k2d, wsf);

  dim3 grid((W + 15) / 16, (H + 63) / 64);
  gauss15_wmma_f32_kernel<<<grid, 32, 0, stream>>>(img, wsf, out, H, W);
}